// LGRUCell_35493609734549
// MI455X (gfx1250) — compile-verified
//
#include <hip/hip_runtime.h>
#include <hip/hip_bf16.h>

// ---------------------------------------------------------------------------
// LGRU cell for MI455X (gfx1250, wave32, WMMA).
// Matrix math in V_WMMA_F32_16X16X4_F32 (f32 in / f32 acc) to preserve the
// reference's float32 numerics through the 128-step recurrence.
//
// Round-2 changes vs round 1:
//  * LDS B-tiles stored K-quad-swizzled: elem(k,c) -> sB[(k>>2)*256 + c*4 + (k&3)]
//    so each WMMA B fragment is one aligned ds_load_b64 (no v_mov repacking).
//  * Double-buffered LDS staging (2 x 8KB) with register prefetch: next 32-K
//    chunk is fetched from global while the current chunk is consumed by WMMA;
//    one workgroup barrier per 32-K chunk.
// ---------------------------------------------------------------------------

typedef __attribute__((ext_vector_type(2))) float v2f;
typedef __attribute__((ext_vector_type(8))) float v8f;

// Problem dimensions (from reference setup_inputs)
constexpr int B_  = 256;
constexpr int T_  = 128;
constexpr int D_  = 256;
constexpr int U_  = 512;
constexpr int BT_ = B_ * T_;     // 32768

constexpr int KCHUNK = 32;       // K per LDS staging round (16KB total LDS)

__device__ __forceinline__ float hsig(float x) {
    return fminf(fmaxf(0.2f * x + 0.5f, 0.0f), 1.0f);
}

// ---------------------------------------------------------------------------
// f32 WMMA GEMM:  C[M,N] = A[M,K] * B[K,N] + bias[N]
//   - 256 threads / block = 8 waves (wave32); wave computes 16(M) x 64(N)
//   - grid = (N/64, M/128); M,N multiples of 128/64; K multiple of 32 (true here)
//
// WMMA f32 16x16x4 fragment layouts (ISA 7.12.2):
//   A (16x4): lane&15 = M row; lane>>4 selects K pair {0,1}/{2,3};
//             VGPR0=K=kb, VGPR1=K=kb+1  -> one float2 global load
//   B (4x16): lane&15 = N col; same K-pair striping -> one ds_load_b64 from
//             the swizzled LDS tile
//   C/D:      VGPR v: rows v (lanes 0-15) / v+8 (lanes 16-31), col = lane&15
// ---------------------------------------------------------------------------
__global__ __launch_bounds__(256)
void gemm_wmma_f32(const float* __restrict__ A, const float* __restrict__ Bm,
                   const float* __restrict__ bias, float* __restrict__ C,
                   int K, int lda, int ldb, int ldc)
{
    // Swizzled B staging: elem(k, c) at sB[buf][(k>>2)*256 + c*4 + (k&3)]
    __shared__ float sB[2][KCHUNK * 64];

    const int tid  = threadIdx.x;
    const int wave = tid >> 5;
    const int lane = tid & 31;
    const int m    = lane & 15;            // M row (A/C) or N col (B/C) in tile
    const int kb   = (lane >> 4) << 1;     // K pair base: 0 or 2

    const int tile_m = (blockIdx.y * 8 + wave) * 16;
    const int tile_n = blockIdx.x * 64;

    v8f acc[4] = {};

    const float* __restrict__ Arow = A + (size_t)(tile_m + m) * lda;

    // Staging slots: slot = tid + s*256 -> q = slot>>6 (K-quad), c = slot&63 (col)
    const int q0 = tid >> 6;   // 0..3
    const int c0 = tid & 63;

    float4 pre[2];

    auto fetch = [&](int k0) {              // global -> registers (4 rows of one col)
        #pragma unroll
        for (int s = 0; s < 2; ++s) {
            const int q = q0 + s * 4;
            const float* src = Bm + (size_t)(k0 + q * 4) * ldb + tile_n + c0;
            pre[s].x = src[0];
            pre[s].y = src[(size_t)ldb];
            pre[s].z = src[(size_t)2 * ldb];
            pre[s].w = src[(size_t)3 * ldb];
        }
    };
    auto stage = [&](int buf) {             // registers -> swizzled LDS (b128 store)
        #pragma unroll
        for (int s = 0; s < 2; ++s) {
            const int q = q0 + s * 4;
            *(float4*)&sB[buf][q * 256 + c0 * 4] = pre[s];
        }
    };

    fetch(0);
    stage(0);
    __syncthreads();

    int buf = 0;
    for (int k0 = 0; k0 < K; k0 += KCHUNK) {
        const bool more = (k0 + KCHUNK) < K;
        if (more) fetch(k0 + KCHUNK);       // overlap next-chunk fetch with WMMA

        #pragma unroll
        for (int kk = 0; kk < KCHUNK / 4; ++kk) {
            const float2 av = *(const float2*)(Arow + k0 + kk * 4 + kb);
            v2f a; a.x = av.x; a.y = av.y;

            #pragma unroll
            for (int j = 0; j < 4; ++j) {
                // contiguous K-pair in swizzled LDS -> single ds_load_b64
                const float2 bv =
                    *(const float2*)&sB[buf][kk * 256 + (j * 16 + m) * 4 + kb];
                v2f b; b.x = bv.x; b.y = bv.y;
                // (neg_a, A, neg_b, B, c_mod, C, reuse_a, reuse_b)
                acc[j] = __builtin_amdgcn_wmma_f32_16x16x4_f32(
                    false, a, false, b, (short)0, acc[j], false, false);
            }
        }

        if (more) stage(buf ^ 1);           // write OTHER buffer: no race with readers
        __syncthreads();                    // one barrier per 32-K chunk
        buf ^= 1;
    }

    // Epilogue: bias add + store (per C/D VGPR layout)
    const int rhalf = (lane >> 4) * 8;
    #pragma unroll
    for (int j = 0; j < 4; ++j) {
        const int col = tile_n + j * 16 + m;
        const float bb = bias[col];
        #pragma unroll
        for (int v = 0; v < 8; ++v) {
            const int row = tile_m + rhalf + v;
            C[(size_t)row * ldc + col] = acc[j][v] + bb;
        }
    }
}

// ---------------------------------------------------------------------------
// Element-wise gate kernels (one thread per (b,u) element; B*U = 131072)
// ---------------------------------------------------------------------------

// z/r/hh gates: h1 = z*h + (1-z)*tanh(xh + r*hhr)
__global__ __launch_bounds__(256)
void gates_kernel(const float* __restrict__ xg, const float* __restrict__ hg,
                  const float* __restrict__ h, float* __restrict__ h1, int t)
{
    const int i = blockIdx.x * blockDim.x + threadIdx.x;
    const int b = i >> 9;          // / U_
    const int u = i & (U_ - 1);    // % U_
    const float* __restrict__ xrow = xg + (size_t)(b * T_ + t) * (5 * U_);
    const float* __restrict__ hrow = hg + (size_t)b * (4 * U_);

    const float xz = xrow[u],          hz  = hrow[u];
    const float xr = xrow[U_ + u],     hr  = hrow[U_ + u];
    const float xh = xrow[2 * U_ + u], hhr = hrow[2 * U_ + u];

    const float z  = hsig(xz + hz);
    const float r  = hsig(xr + hr);
    const float hh = tanhf(xh + r * hhr);
    h1[i] = z * h[i] + (1.0f - z) * hh;
}

// inner t-layer: h1 = zt*h1 + (1-zt)*tanh(rt*th)
__global__ __launch_bounds__(256)
void tlayer_kernel(const float* __restrict__ tg, float* __restrict__ h1)
{
    const int i = blockIdx.x * blockDim.x + threadIdx.x;
    const int b = i >> 9;
    const int u = i & (U_ - 1);
    const float* __restrict__ trow = tg + (size_t)b * (3 * U_);

    const float zt = hsig(trow[u]);
    const float rt = hsig(trow[U_ + u]);
    const float ht = tanhf(rt * trow[2 * U_ + u]);
    h1[i] = zt * h1[i] + (1.0f - zt) * ht;
}

// output gate: h_out = l*h1 + (1-l)*tanh(xl2); writes ys[:,t,:] and carries h
__global__ __launch_bounds__(256)
void out_kernel(const float* __restrict__ xg, const float* __restrict__ hg,
                const float* __restrict__ h1, float* __restrict__ h,
                float* __restrict__ ys, int t)
{
    const int i = blockIdx.x * blockDim.x + threadIdx.x;
    const int b = i >> 9;
    const int u = i & (U_ - 1);
    const float* __restrict__ xrow = xg + (size_t)(b * T_ + t) * (5 * U_);

    const float xl  = xrow[3 * U_ + u];
    const float xl2 = xrow[4 * U_ + u];
    const float hl  = hg[(size_t)b * (4 * U_) + 3 * U_ + u];

    const float l  = hsig(xl + hl);
    const float ho = l * h1[i] + (1.0f - l) * tanhf(xl2);
    h[i] = ho;
    ys[((size_t)b * T_ + t) * U_ + u] = ho;
}

// ---------------------------------------------------------------------------
// Host-side launch (graph-capture safe: only kernel launches + async d2d copy)
// ---------------------------------------------------------------------------
extern "C" void kernel_launch(void* const* d_in, const int* in_sizes, int n_in,
                              void* d_out, int out_size, void* d_ws, size_t ws_size,
                              hipStream_t stream)
{
    const float* x     = (const float*)d_in[0];   // (B,T,D)
    const float* h0    = (const float*)d_in[1];   // (B,U)
    const float* kern  = (const float*)d_in[2];   // (D,5U)
    const float* rker  = (const float*)d_in[3];   // (U,4U)
    const float* tker  = (const float*)d_in[4];   // (2,U,3U)
    const float* bias  = (const float*)d_in[5];   // (2,5U)
    const float* tbias = (const float*)d_in[6];   // (2,3U)
    float* ys          = (float*)d_out;           // (B,T,U)

    // Workspace layout (floats): ~340 MB total
    float* ws = (float*)d_ws;
    size_t off = 0;
    float* xg = ws + off; off += (size_t)BT_ * 5 * U_;   // 83,886,080
    float* h  = ws + off; off += (size_t)B_ * U_;
    float* h1 = ws + off; off += (size_t)B_ * U_;
    float* hg = ws + off; off += (size_t)B_ * 4 * U_;
    float* tg = ws + off; off += (size_t)B_ * 3 * U_;
    (void)off; (void)ws_size; (void)in_sizes; (void)n_in; (void)out_size;

    // h = h0
    hipMemcpyAsync(h, h0, (size_t)B_ * U_ * sizeof(float),
                   hipMemcpyDeviceToDevice, stream);

    const dim3 blk(256);

    // Phase 1: xg_all = x @ kernel + bias[0]   (32768 x 256 x 2560)
    gemm_wmma_f32<<<dim3(5 * U_ / 64, BT_ / 128), blk, 0, stream>>>(
        x, kern, bias, xg, /*K=*/D_, /*lda=*/D_, /*ldb=*/5 * U_, /*ldc=*/5 * U_);

    // Phase 2: sequential scan over T
    const int ew_grid = (B_ * U_) / 256;   // 512 blocks
    for (int t = 0; t < T_; ++t) {
        // hg = h @ recurrent_kernel + bias[1][:4U]   (256 x 512 x 2048)
        gemm_wmma_f32<<<dim3(4 * U_ / 64, B_ / 128), blk, 0, stream>>>(
            h, rker, bias + 5 * U_, hg, /*K=*/U_, U_, 4 * U_, 4 * U_);

        gates_kernel<<<ew_grid, blk, 0, stream>>>(xg, hg, h, h1, t);

        for (int i = 0; i < 2; ++i) {
            // tg = h1 @ t_kernels[i] + t_bias[i]     (256 x 512 x 1536)
            gemm_wmma_f32<<<dim3(3 * U_ / 64, B_ / 128), blk, 0, stream>>>(
                h1, tker + (size_t)i * U_ * 3 * U_, tbias + (size_t)i * 3 * U_,
                tg, /*K=*/U_, U_, 3 * U_, 3 * U_);

            tlayer_kernel<<<ew_grid, blk, 0, stream>>>(tg, h1);
        }

        out_kernel<<<ew_grid, blk, 0, stream>>>(xg, hg, h1, h, ys, t);
    }
}